// TopkRouter_3521873183481
// MI455X (gfx1250) — compile-verified
//
#include <hip/hip_runtime.h>
#include <hip/hip_bf16.h>
#include <cmath>

// MoE top-2 router, fused. Shapes: B=8, N=4096, D=1024, E=8, K=2.
// d_out layout (float): dispatch [B,N,4] | combine [B,N,2] | aux_loss | z_loss

typedef float v2f __attribute__((ext_vector_type(2)));
typedef float v8f __attribute__((ext_vector_type(8)));
typedef int   v4i __attribute__((vector_size(16)));   // matches builtin proto

#define B_DIM 8
#define N_TOK 4096
#define D_DIM 1024
#define E_DIM 8
#define K_TOP 2
#define LDS_STRIDE 1032  // 1024 + 8 pad -> conflict-free ds_load_b64 across lanes
#define WROWS 9          // 8 expert rows + 1 shared zero row (expert padding)
#define KC 32            // k-chunk staged per async group (floats)
#define TS 36            // X-tile row stride: 32 + 4 pad (banks 4*lm distinct)

#if defined(__HIP_DEVICE_COMPILE__) && \
    __has_builtin(__builtin_amdgcn_global_load_async_to_lds_b128) && \
    __has_builtin(__builtin_amdgcn_s_wait_asynccnt)
#define HAS_ASYNC 1
#else
#define HAS_ASYNC 0
#endif

// ---------------------------------------------------------------------------
// Kernel 0: zero the small global accumulators (ws is poisoned, not re-zeroed)
// ---------------------------------------------------------------------------
__global__ void init_accum_kernel(float* g_sumP, float* g_cnt, float* g_z) {
    int t = threadIdx.x;
    if (t < B_DIM * E_DIM) { g_sumP[t] = 0.f; g_cnt[t] = 0.f; }
    if (t == B_DIM * E_DIM) *g_z = 0.f;
}

// ---------------------------------------------------------------------------
// Kernel 1: streaming GEMM (WMMA f32 16x16x4) + double softmax + top-2 +
//           partial loss accumulation. 128 threads = 4 waves, 16 tokens/wave.
// logits^T = W^T (16x4, experts padded via zero row) x X^T (4x16 tokens).
// D layout (ISA 7.12.2): VGPR r, lanes 0-15 -> M=r (expert), N=lane (token),
// so lane t<16 ends holding token t's 8 logits in c[0..7].
// X is staged Global->LDS with async b128 copies (512B, 4 full cache lines per
// request), double-buffered against WMMA via s_wait_asynccnt.
// ---------------------------------------------------------------------------
__global__ __launch_bounds__(128) void router_fused_kernel(
    const float* __restrict__ X, const float* __restrict__ Wg,
    float* __restrict__ disp, int* __restrict__ idx_ws, float* __restrict__ wts_ws,
    float* __restrict__ g_sumP, float* __restrict__ g_cnt, float* __restrict__ g_z)
{
    __shared__ float lds_wt[WROWS * LDS_STRIDE];   // W transposed + zero row
    __shared__ float lds_x[4][2][16 * TS];         // per-wave double buffers
    __shared__ float lds_sumP[E_DIM];
    __shared__ float lds_cnt[E_DIM];
    __shared__ float lds_z;

    const int tid = threadIdx.x;
    if (tid < E_DIM) { lds_sumP[tid] = 0.f; lds_cnt[tid] = 0.f; }
    if (tid == 0) lds_z = 0.f;
    for (int i = tid; i < LDS_STRIDE; i += 128)    // zero row (expert pad)
        lds_wt[8 * LDS_STRIDE + i] = 0.f;
    for (int i = tid; i < D_DIM * E_DIM; i += 128) {
        int d = i >> 3, e = i & 7;
        lds_wt[e * LDS_STRIDE + d] = Wg[i];        // W[d][e] -> lds_wt[e][d]
    }
    __syncthreads();

    const int wave = tid >> 5;
    const int lane = tid & 31;
    const int hl   = lane >> 4;   // 0: K pair {0,1}, 1: K pair {2,3}
    const int lm   = lane & 15;   // A: expert row (M); B: token col (N)
    const long tokBase = ((long)blockIdx.x * 4 + wave) * 16;

    v8f c = {0.f, 0.f, 0.f, 0.f, 0.f, 0.f, 0.f, 0.f};
    // lanes lm>=8 read the shared zero row (same addr -> LDS broadcast)
    const float* aptr = lds_wt + ((lm < E_DIM) ? lm : E_DIM) * LDS_STRIDE + 2 * hl;

#if HAS_ASYNC
    // stage geometry: instr i covers rows 4i..4i+3, 128B per row (KC=32 floats)
    const int sr = lane >> 3;        // row within 4-row group
    const int sc = (lane & 7) * 4;   // float offset within row
    float* buf0 = &lds_x[wave][0][0];
    float* buf1 = &lds_x[wave][1][0];

    auto stage = [&](float* buf, int c0) {
        #pragma unroll
        for (int i = 0; i < 4; ++i) {
            int r = 4 * i + sr;
            const float* g = X + (size_t)(tokBase + r) * D_DIM + c0 + sc;
            float* l = buf + r * TS + sc;
            __builtin_amdgcn_global_load_async_to_lds_b128(
                (v4i*)g,
                (__attribute__((address_space(3))) v4i*)l, 0, 0);
        }
    };
    auto consume = [&](const float* buf, int kglob) {
        #pragma unroll
        for (int kc = 0; kc < KC; kc += 4) {
            v2f a  = *(const v2f*)(aptr + kglob + kc);
            v2f bv = *(const v2f*)(buf + lm * TS + kc + 2 * hl);
            c = __builtin_amdgcn_wmma_f32_16x16x4_f32(false, a, false, bv,
                                                      (short)0, c, false, false);
        }
    };

    stage(buf0, 0);
    #pragma unroll 1
    for (int ch = 0; ch < D_DIM / KC; ++ch) {
        float* cur = (ch & 1) ? buf1 : buf0;
        float* nxt = (ch & 1) ? buf0 : buf1;
        if (ch + 1 < D_DIM / KC) {
            stage(nxt, (ch + 1) * KC);
            __builtin_amdgcn_s_wait_asynccnt(4);   // previous group landed
        } else {
            __builtin_amdgcn_s_wait_asynccnt(0);
        }
        consume(cur, ch * KC);
    }
#else
    // fallback: round-1 direct global b64 streaming path (known-good codegen)
    {
        const float* xrow = X + (size_t)(tokBase + lm) * D_DIM + 2 * hl;
        #pragma unroll 4
        for (int k = 0; k < D_DIM; k += 4) {
            v2f a  = *(const v2f*)(aptr + k);
            v2f bv = *(const v2f*)(xrow + k);
            c = __builtin_amdgcn_wmma_f32_16x16x4_f32(false, a, false, bv,
                                                      (short)0, c, false, false);
        }
    }
#endif

    if (hl == 0) {  // lanes 0-15: one token each
        const long g = tokBase + lm;          // flat token id
        float l[E_DIM];
        #pragma unroll
        for (int e = 0; e < E_DIM; ++e) l[e] = c[e];

        // logsumexp -> z loss term
        float mx = l[0];
        #pragma unroll
        for (int e = 1; e < E_DIM; ++e) mx = fmaxf(mx, l[e]);
        float s = 0.f;
        #pragma unroll
        for (int e = 0; e < E_DIM; ++e) s += __expf(l[e] - mx);
        float lse = mx + __logf(s);

        // softmax #1
        float p1[E_DIM];
        #pragma unroll
        for (int e = 0; e < E_DIM; ++e) p1[e] = __expf(l[e] - mx) / s;
        // softmax #2 (faithful to reference's double softmax)
        float mx2 = p1[0];
        #pragma unroll
        for (int e = 1; e < E_DIM; ++e) mx2 = fmaxf(mx2, p1[e]);
        float s2 = 0.f;
        #pragma unroll
        for (int e = 0; e < E_DIM; ++e) s2 += __expf(p1[e] - mx2);
        float p2[E_DIM];
        #pragma unroll
        for (int e = 0; e < E_DIM; ++e) p2[e] = __expf(p1[e] - mx2) / s2;

        // top-2 (earliest index wins ties, like jax top_k)
        float v0 = -1e30f, v1 = -1e30f; int i0 = 0, i1 = 0;
        #pragma unroll
        for (int e = 0; e < E_DIM; ++e) {
            float v = p2[e];
            if (v > v0)      { v1 = v0; i1 = i0; v0 = v; i0 = e; }
            else if (v > v1) { v1 = v;  i1 = e; }
        }

        disp[g * 4 + 0] = (float)i0;
        disp[g * 4 + 1] = (float)i1;
        idx_ws[g * 2 + 0] = i0;  idx_ws[g * 2 + 1] = i1;
        wts_ws[g * 2 + 0] = v0;  wts_ws[g * 2 + 1] = v1;

        atomicAdd(&lds_z, lse * lse);
        atomicAdd(&lds_cnt[i0], 1.f);
        atomicAdd(&lds_cnt[i1], 1.f);
        #pragma unroll
        for (int e = 0; e < E_DIM; ++e) atomicAdd(&lds_sumP[e], p2[e]);
    }
    __syncthreads();

    // block covers 64 tokens, all in one batch (4096 % 64 == 0)
    const int bb = (int)(((long)blockIdx.x * 64) / N_TOK);
    if (tid < E_DIM)            atomicAdd(&g_sumP[bb * E_DIM + tid], lds_sumP[tid]);
    else if (tid < 2 * E_DIM)   atomicAdd(&g_cnt[bb * E_DIM + tid - E_DIM], lds_cnt[tid - E_DIM]);
    else if (tid == 2 * E_DIM)  atomicAdd(g_z, lds_z);
}

// ---------------------------------------------------------------------------
// Kernel 2: per-(batch, expert) priority cumsum over flat (k, token) order.
// One wave per pair; ballot + popcount prefix gives within-expert rank.
// ---------------------------------------------------------------------------
__global__ __launch_bounds__(32) void rank_kernel(
    const int* __restrict__ idx_ws, const float* __restrict__ wts_ws,
    const int* __restrict__ capPtr,
    float* __restrict__ disp, float* __restrict__ comb)
{
    const int b = blockIdx.x >> 3;
    const int e = blockIdx.x & 7;
    const int cap = *capPtr;
    const int lane = threadIdx.x;
    const unsigned ltmask = (1u << lane) - 1u;

    int running = 0;
    for (int i0 = 0; i0 < K_TOP * N_TOK; i0 += 32) {
        int i = i0 + lane;
        int k = i >> 12;        // i / N_TOK  (k-major flatten)
        int n = i & (N_TOK - 1);
        long g = (long)b * N_TOK + n;
        bool match = (idx_ws[g * 2 + k] == e);
#if defined(__HIP_DEVICE_COMPILE__) && __has_builtin(__builtin_amdgcn_ballot_w32)
        unsigned m = __builtin_amdgcn_ballot_w32(match);
#else
        unsigned m = (unsigned)__ballot(match);
#endif
        int rank = running + __popc(m & ltmask);
        if (match) {
            disp[g * 4 + 2 + k] = (float)rank;
            comb[g * 2 + k] = wts_ws[g * 2 + k] * ((rank < cap) ? 1.f : 0.f);
        }
        running += __popc(m);
    }
}

// ---------------------------------------------------------------------------
// Kernel 3: finalize aux_loss and router_z_loss (64 (b,e) pairs).
// ---------------------------------------------------------------------------
__global__ __launch_bounds__(64) void finalize_kernel(
    const float* __restrict__ g_sumP, const float* __restrict__ g_cnt,
    const float* __restrict__ g_z, float* __restrict__ outAux, float* __restrict__ outZ)
{
    __shared__ float red[64];
    int t = threadIdx.x;
    float density  = g_cnt[t]  / (float)N_TOK;
    float dproxy   = g_sumP[t] / (float)N_TOK;
    red[t] = density * dproxy;
    __syncthreads();
    for (int s = 32; s > 0; s >>= 1) {
        if (t < s) red[t] += red[t + s];
        __syncthreads();
    }
    if (t == 0) {
        *outAux = red[0] / (float)(B_DIM * E_DIM) * (float)(E_DIM * E_DIM);
        *outZ   = (*g_z) / (float)(B_DIM * N_TOK);
    }
}

// ---------------------------------------------------------------------------
extern "C" void kernel_launch(void* const* d_in, const int* in_sizes, int n_in,
                              void* d_out, int out_size, void* d_ws, size_t ws_size,
                              hipStream_t stream)
{
    const float* X   = (const float*)d_in[0];   // [B,N,D]
    const float* Wg  = (const float*)d_in[1];   // [D,E]
    const int*   cap = (const int*)d_in[2];     // scalar

    float* out  = (float*)d_out;
    float* disp = out;                              // [B,N,4]
    float* comb = out + (size_t)B_DIM * N_TOK * 4;  // [B,N,2]
    float* aux  = out + (size_t)B_DIM * N_TOK * 6;  // scalar
    float* zl   = aux + 1;                          // scalar

    char*  ws     = (char*)d_ws;
    int*   idx_ws = (int*)ws;                                        // B*N*2 ints
    float* wts_ws = (float*)(ws + (size_t)B_DIM * N_TOK * 2 * 4);    // B*N*2 floats
    float* g_sumP = (float*)(ws + (size_t)B_DIM * N_TOK * 4 * 4);    // 64
    float* g_cnt  = g_sumP + B_DIM * E_DIM;                          // 64
    float* g_z    = g_cnt  + B_DIM * E_DIM;                          // 1

    init_accum_kernel<<<1, 96, 0, stream>>>(g_sumP, g_cnt, g_z);
    router_fused_kernel<<<(B_DIM * N_TOK) / 64, 128, 0, stream>>>(
        X, Wg, disp, idx_ws, wts_ws, g_sumP, g_cnt, g_z);
    rank_kernel<<<B_DIM * E_DIM, 32, 0, stream>>>(idx_ws, wts_ws, cap, disp, comb);
    finalize_kernel<<<1, 64, 0, stream>>>(g_sumP, g_cnt, g_z, aux, zl);
}